// Net_36189394437011
// MI455X (gfx1250) — compile-verified
//
#include <hip/hip_runtime.h>
#include <math.h>

typedef __attribute__((ext_vector_type(2))) float v2f;
typedef __attribute__((ext_vector_type(8))) float v8f;
typedef __attribute__((ext_vector_type(4))) unsigned int v4u;
typedef __attribute__((ext_vector_type(8))) int v8i;
typedef __attribute__((ext_vector_type(4))) int v4i;

#define FIN  500
#define FH   64
#define FOUT 5
#define BM   128            // rows per block (4 waves x 32)
#define BK   100            // K-slab (500 = 5 slabs)
#define WPITCH 160          // interleaved-W pitch in floats (=32 mod 64 banks -> conflict free)
#define XBUF (BM * BK)      // floats per x slab buffer (12800)
#define SW_FLOATS ((BK / 2) * WPITCH)  // 8000
#define SMEM_FLOATS (2 * XBUF + SW_FLOATS)

#if defined(__has_builtin)
#if __has_builtin(__builtin_amdgcn_tensor_load_to_lds) && __has_builtin(__builtin_amdgcn_s_wait_tensorcnt)
#define HAVE_TDM 1
#endif
#endif
#ifndef HAVE_TDM
#define HAVE_TDM 0
#endif

// ---------------- utility kernels ----------------

__global__ void fill_zero(float* __restrict__ p, long n) {
  long i = (long)blockIdx.x * blockDim.x + threadIdx.x;
  if (i < n) p[i] = 0.0f;
}

__global__ void deg_kernel(const int* __restrict__ dst, float* __restrict__ deg, long E) {
  long e = (long)blockIdx.x * blockDim.x + threadIdx.x;
  if (e < E) atomicAdd(&deg[dst[e]], 1.0f);
}

// deg -> dinv in place; +1 accounts for the self-loop on every node
__global__ void dinv_kernel(float* __restrict__ deg, int N) {
  int i = blockIdx.x * blockDim.x + threadIdx.x;
  if (i < N) deg[i] = 1.0f / sqrtf(deg[i] + 1.0f);
}

// ---------------- TDM issue: 2D tile (rows x BK cols) of x -> LDS ----------------

#if HAVE_TDM
__device__ __forceinline__ void tdm_load_xtile(const float* g, unsigned ldsOff, unsigned rowsRem) {
  unsigned long long ga = (unsigned long long)(const void*)g;
  v4u g0;
  g0.x = 1u;                                       // count=1, user descriptor
  g0.y = ldsOff;                                   // lds_addr (bytes)
  g0.z = (unsigned)ga;                             // global_addr[31:0]
  g0.w = (unsigned)((ga >> 32) & 0x01FFFFFFu)      // global_addr[56:32]
         | (2u << 30);                             // type = 2 ("image")
  v8i g1;
  g1[0] = (int)(2u << 16);                         // data_size = 4 bytes; wg_mask=0
  g1[1] = (int)((unsigned)(FIN & 0xFFFF) << 16);   // tensor_dim0[15:0]  (=500)
  g1[2] = (int)(((unsigned)FIN >> 16) | ((rowsRem & 0xFFFFu) << 16));  // td0 hi | td1 lo
  g1[3] = (int)(((rowsRem >> 16) & 0xFFFFu) | ((unsigned)BK << 16));   // td1 hi | tile_dim0
  g1[4] = (int)BM;                                 // tile_dim1 = 128 rows; tile_dim2 = 0
  g1[5] = (int)FIN;                                // tensor_dim0_stride[31:0] (=500)
  g1[6] = 0;
  g1[7] = 0;
  v4i z4 = {0, 0, 0, 0};
  v8i z8 = {0, 0, 0, 0, 0, 0, 0, 0};
  __builtin_amdgcn_tensor_load_to_lds(g0, g1, z4, z4, z8, 0);
}
#endif

// ---------------- GEMM1: h = x @ W1 via fp32 WMMA ----------------
// Block: 128 threads = 4 waves; each wave computes 32 rows x 64 cols.
// x tiles staged to LDS by the Tensor Data Mover (double buffered, TENSORcnt);
// W1 staged K-pair-interleaved so each B fragment is one conflict-free ds_load_b64.

__global__ __launch_bounds__(128) void gemm1_wmma(const float* __restrict__ x,
                                                  const float* __restrict__ W,
                                                  float* __restrict__ h, int N) {
  extern __shared__ __align__(16) float smem[];
  float* sWl = smem + 2 * XBUF;

  const int tid  = threadIdx.x;
  const int lane = tid & 31;
  const int wave = tid >> 5;      // 0..3
  const int lr   = lane & 15;     // 0..15
  const int hi   = lane >> 4;     // 0 or 1
  const long rowBlk = (long)blockIdx.x * BM;
  const unsigned rowsRem = (unsigned)(N - rowBlk);   // grid guarantees >= 1

  v8f acc[2][4] = {};

  const int S = FIN / BK;  // 5 slabs

#if HAVE_TDM
  if (wave == 0) {
    tdm_load_xtile(x + rowBlk * FIN, 0u, rowsRem);
    __builtin_amdgcn_s_wait_tensorcnt(0);
  }
#endif

  for (int s = 0; s < S; ++s) {
    __syncthreads();   // xbuf[s&1] published; sW free
#if HAVE_TDM
    if (wave == 0 && s + 1 < S)
      tdm_load_xtile(x + rowBlk * FIN + (s + 1) * BK,
                     (unsigned)(((s + 1) & 1) * XBUF * 4), rowsRem);
#else
    {  // synchronous x staging (single buffer) with zero-fill tail
      float* xb0 = smem;
      for (int i = tid; i < (BM * BK) / 4; i += 128) {
        int rr = i / (BK / 4);
        int cc = (i % (BK / 4)) * 4;
        long grow = rowBlk + rr;
        float4 v = {0.f, 0.f, 0.f, 0.f};
        if (grow < N) v = *reinterpret_cast<const float4*>(x + grow * FIN + s * BK + cc);
        *reinterpret_cast<float4*>(xb0 + rr * BK + cc) = v;
      }
    }
#endif
    // stage W slab s, K-pair interleaved: sW[(k/2)*WPITCH + c*2 + (k&1)] = W[s*BK+k][c]
    for (int i = tid; i < BK * FH; i += 128) {
      int k = i >> 6;
      int c = i & (FH - 1);
      sWl[(k >> 1) * WPITCH + c * 2 + (k & 1)] = W[(s * BK + k) * FH + c];
    }
    __syncthreads();   // sW (and fallback xbuf) ready

#if HAVE_TDM
    const float* xb = smem + (s & 1) * XBUF;
#else
    const float* xb = smem;
#endif
    const float* xrow0 = xb + (wave * 32 + lr) * BK;   // M-subtile 0
    const float* xrow1 = xrow0 + 16 * BK;              // M-subtile 1

    #pragma unroll 5
    for (int k = 0; k < BK; k += 4) {
      const int kb = k + 2 * hi;  // ISA A layout: lanes 0-15 -> K+{0,1}, 16-31 -> K+{2,3}
      v2f a0 = *reinterpret_cast<const v2f*>(xrow0 + kb);
      v2f a1 = *reinterpret_cast<const v2f*>(xrow1 + kb);
      const float* wp = sWl + (k / 2 + hi) * WPITCH + 2 * lr;
      v2f b0 = *reinterpret_cast<const v2f*>(wp);
      v2f b1 = *reinterpret_cast<const v2f*>(wp + 32);
      v2f b2 = *reinterpret_cast<const v2f*>(wp + 64);
      v2f b3 = *reinterpret_cast<const v2f*>(wp + 96);
      acc[0][0] = __builtin_amdgcn_wmma_f32_16x16x4_f32(false, a0, false, b0, (short)0, acc[0][0], false, false);
      acc[0][1] = __builtin_amdgcn_wmma_f32_16x16x4_f32(false, a0, false, b1, (short)0, acc[0][1], false, false);
      acc[0][2] = __builtin_amdgcn_wmma_f32_16x16x4_f32(false, a0, false, b2, (short)0, acc[0][2], false, false);
      acc[0][3] = __builtin_amdgcn_wmma_f32_16x16x4_f32(false, a0, false, b3, (short)0, acc[0][3], false, false);
      acc[1][0] = __builtin_amdgcn_wmma_f32_16x16x4_f32(false, a1, false, b0, (short)0, acc[1][0], false, false);
      acc[1][1] = __builtin_amdgcn_wmma_f32_16x16x4_f32(false, a1, false, b1, (short)0, acc[1][1], false, false);
      acc[1][2] = __builtin_amdgcn_wmma_f32_16x16x4_f32(false, a1, false, b2, (short)0, acc[1][2], false, false);
      acc[1][3] = __builtin_amdgcn_wmma_f32_16x16x4_f32(false, a1, false, b3, (short)0, acc[1][3], false, false);
    }
    __syncthreads();   // compute done; sW/xbuf reusable
#if HAVE_TDM
    if (wave == 0 && s + 1 < S) __builtin_amdgcn_s_wait_tensorcnt(0);
#endif
  }

  // C layout: VGPR r -> M=r (lanes 0-15) / M=r+8 (lanes 16-31)
  #pragma unroll
  for (int mt = 0; mt < 2; ++mt) {
    #pragma unroll
    for (int r = 0; r < 8; ++r) {
      long row = rowBlk + wave * 32 + mt * 16 + r + 8 * hi;
      if (row < N) {
        float* o = h + row * FH + lr;
        o[0]  = acc[mt][0][r];
        o[16] = acc[mt][1][r];
        o[32] = acc[mt][2][r];
        o[48] = acc[mt][3][r];
      }
    }
  }
}

// ---------------- edge aggregation, 64 features (wave per edge) ----------------

__global__ __launch_bounds__(256) void agg64(const float* __restrict__ h,
                                             const int* __restrict__ src,
                                             const int* __restrict__ dst,
                                             const float* __restrict__ dinv,
                                             float* __restrict__ out, long E, int N) {
  long idx = (long)blockIdx.x * blockDim.x + threadIdx.x;
  long e = idx >> 5;
  int lane = (int)(idx & 31);
  if (e >= E + (long)N) return;
  int s, d;
  if (e < E) { s = src[e]; d = dst[e]; }
  else       { s = d = (int)(e - E); }          // self-loops appended
  const float nrm = dinv[s] * dinv[d];
  const float2 v = *reinterpret_cast<const float2*>(h + (long)s * FH + 2 * lane);
  float* o = out + (long)d * FH + 2 * lane;
  atomicAdd(o,     v.x * nrm);
  atomicAdd(o + 1, v.y * nrm);
}

__global__ void bias_relu(float* __restrict__ a, const float* __restrict__ b, long total) {
  long i = (long)blockIdx.x * blockDim.x + threadIdx.x;
  if (i >= total) return;
  float v = a[i] + b[i & (FH - 1)];
  a[i] = v > 0.0f ? v : 0.0f;
}

// ---------------- GEMM2: h2 = z1 @ W2  ([N,64]x[64,5]) ----------------

__global__ void gemm2(const float* __restrict__ z1, const float* __restrict__ W2,
                      float* __restrict__ h2, int N) {
  int n = blockIdx.x * blockDim.x + threadIdx.x;
  if (n >= N) return;
  const float* row = z1 + (long)n * FH;
  float acc[FOUT] = {0.f, 0.f, 0.f, 0.f, 0.f};
  #pragma unroll 8
  for (int k = 0; k < FH; ++k) {
    float v = row[k];
    #pragma unroll
    for (int c = 0; c < FOUT; ++c) acc[c] = fmaf(v, W2[k * FOUT + c], acc[c]);
  }
  #pragma unroll
  for (int c = 0; c < FOUT; ++c) h2[(long)n * FOUT + c] = acc[c];
}

// ---------------- edge aggregation, 5 features (thread per edge) ----------------

__global__ void agg5(const float* __restrict__ h2, const int* __restrict__ src,
                     const int* __restrict__ dst, const float* __restrict__ dinv,
                     float* __restrict__ out, long E, int N) {
  long e = (long)blockIdx.x * blockDim.x + threadIdx.x;
  if (e >= E + (long)N) return;
  int s, d;
  if (e < E) { s = src[e]; d = dst[e]; }
  else       { s = d = (int)(e - E); }
  const float nrm = dinv[s] * dinv[d];
  const float* hv = h2 + (long)s * FOUT;
  float* o = out + (long)d * FOUT;
  #pragma unroll
  for (int c = 0; c < FOUT; ++c) atomicAdd(o + c, hv[c] * nrm);
}

// ---------------- bias + log_softmax ----------------

__global__ void final_logsoftmax(const float* __restrict__ agg2, const float* __restrict__ b2,
                                 float* __restrict__ out, int N) {
  int n = blockIdx.x * blockDim.x + threadIdx.x;
  if (n >= N) return;
  float v[FOUT];
  float m = -INFINITY;
  #pragma unroll
  for (int c = 0; c < FOUT; ++c) {
    v[c] = agg2[(long)n * FOUT + c] + b2[c];
    m = fmaxf(m, v[c]);
  }
  float ssum = 0.0f;
  #pragma unroll
  for (int c = 0; c < FOUT; ++c) ssum += expf(v[c] - m);
  float l = logf(ssum);
  #pragma unroll
  for (int c = 0; c < FOUT; ++c) out[(long)n * FOUT + c] = v[c] - m - l;
}

// ---------------- launcher ----------------

extern "C" void kernel_launch(void* const* d_in, const int* in_sizes, int n_in,
                              void* d_out, int out_size, void* d_ws, size_t ws_size,
                              hipStream_t stream) {
  const float* x  = (const float*)d_in[0];   // [N,500]
  const int*   ei = (const int*)d_in[1];     // [2,E]
  const float* W1 = (const float*)d_in[2];   // [500,64]
  const float* b1 = (const float*)d_in[3];   // [64]
  const float* W2 = (const float*)d_in[4];   // [64,5]
  const float* b2 = (const float*)d_in[5];   // [5]

  const int  N = in_sizes[0] / FIN;
  const long E = in_sizes[1] / 2;
  const int* srcp = ei;
  const int* dstp = ei + E;

  float* ws   = (float*)d_ws;
  float* dinv = ws;                          // N
  float* hlin = dinv + N;                    // N*64
  float* agg1 = hlin + (long)N * FH;         // N*64
  float* h2   = agg1 + (long)N * FH;         // N*5
  float* agg2 = h2 + (long)N * FOUT;         // N*5

  const long NF = (long)N * FH;
  const long NO = (long)N * FOUT;
  const long ET = E + (long)N;               // edges + self-loops

  fill_zero<<<(unsigned)((N + 255) / 256), 256, 0, stream>>>(dinv, (long)N);
  fill_zero<<<(unsigned)((NF + 255) / 256), 256, 0, stream>>>(agg1, NF);
  fill_zero<<<(unsigned)((NO + 255) / 256), 256, 0, stream>>>(agg2, NO);

  deg_kernel<<<(unsigned)((E + 255) / 256), 256, 0, stream>>>(dstp, dinv, E);
  dinv_kernel<<<(unsigned)((N + 255) / 256), 256, 0, stream>>>(dinv, N);

  gemm1_wmma<<<(unsigned)((N + BM - 1) / BM), 128, SMEM_FLOATS * sizeof(float), stream>>>(
      x, W1, hlin, N);

  agg64<<<(unsigned)((ET * 32 + 255) / 256), 256, 0, stream>>>(hlin, srcp, dstp, dinv, agg1, E, N);
  bias_relu<<<(unsigned)((NF + 255) / 256), 256, 0, stream>>>(agg1, b1, NF);

  gemm2<<<(unsigned)((N + 255) / 256), 256, 0, stream>>>(agg1, W2, h2, N);
  agg5<<<(unsigned)((ET + 255) / 256), 256, 0, stream>>>(h2, srcp, dstp, dinv, agg2, E, N);

  final_logsoftmax<<<(unsigned)((N + 255) / 256), 256, 0, stream>>>(agg2, b2, (float*)d_out, N);
}